// FirstDelayActionPredict_VAD_88622355185707
// MI455X (gfx1250) — compile-verified
//
#include <hip/hip_runtime.h>
#include <math.h>

// ---------------------------------------------------------------------------
// FirstDelayActionPredict_VAD: batch-1 LSTM step on MI455X (gfx1250)
// Bandwidth-bound fp32 matvecs done as V_WMMA_F32_16X16X4_F32 tiles.
// ---------------------------------------------------------------------------

typedef float v2f __attribute__((ext_vector_type(2)));
typedef float v4f __attribute__((ext_vector_type(4)));
typedef float v8f __attribute__((ext_vector_type(8)));

#define IN_DIM 8192
#define H_DIM  4096
#define H4_DIM 16384

// workspace layout (floats)
#define ZPART_OFF 0        // [4][H_DIM]   fc1 partial sums per K-chunk
#define GPART_OFF 16384    // [4][H4_DIM]  gate partials: slot = mat*2 + kc
#define ZBUF_OFF  81920    // [H_DIM]      z = relu(fc1 + b1)
// total = 86016 floats = 336 KB

// ---- 4 chained f32 WMMAs covering 16 K-values ------------------------------
// Per the ISA A/B layouts: lanes 0-15 supply the first half of each WMMA's
// K-set, lanes 16-31 the second half. We permute the K assignment so each
// lane streams two contiguous float4s from its weight row (coalesced b128).
__device__ __forceinline__ v8f wmma16k(v8f acc, v4f w0, v4f w1, v4f x0, v4f x1) {
    v2f a0 = {w0.x, w0.y}, a1 = {w0.z, w0.w};
    v2f a2 = {w1.x, w1.y}, a3 = {w1.z, w1.w};
    v2f b0 = {x0.x, x0.y}, b1 = {x0.z, x0.w};
    v2f b2 = {x1.x, x1.y}, b3 = {x1.z, x1.w};
    acc = __builtin_amdgcn_wmma_f32_16x16x4_f32(false, a0, false, b0, (short)0, acc, false, false);
    acc = __builtin_amdgcn_wmma_f32_16x16x4_f32(false, a1, false, b1, (short)0, acc, false, false);
    acc = __builtin_amdgcn_wmma_f32_16x16x4_f32(false, a2, false, b2, (short)0, acc, false, false);
    acc = __builtin_amdgcn_wmma_f32_16x16x4_f32(false, a3, false, b3, (short)0, acc, false, false);
    return acc;
}

// One wave computes 16 rows x 2048 K of W * vec and writes 16 partial sums.
__device__ __forceinline__ void matvec_tile_wave(
    const float* __restrict__ W, int ld, int koff,
    const float* xs /*LDS, 2048 floats*/, int row_base,
    float* __restrict__ outp)
{
    const int lane = threadIdx.x & 31;
    const int m    = lane & 15;   // row within tile
    const int hi   = lane >> 4;   // 0: K 0..7 of block, 1: K 8..15
    const float* wrow = W + (size_t)(row_base + m) * (size_t)ld + (size_t)(koff + 8 * hi);
    const float* xrow = xs + 8 * hi;

    v8f acc = {};
#pragma unroll 4
    for (int k = 0; k < 2048; k += 16) {
        // streamed-once weights: nontemporal b128 loads
        v4f w0 = __builtin_nontemporal_load((const v4f*)(wrow + k));
        v4f w1 = __builtin_nontemporal_load((const v4f*)(wrow + k + 4));
        v4f x0 = *(const v4f*)(xrow + k);
        v4f x1 = *(const v4f*)(xrow + k + 4);
        acc = wmma16k(acc, w0, w1, x0, x1);
    }

    // C layout: VGPR r, lanes 0-15 -> M=r, lanes 16-31 -> M=8+r; N = lane%16.
    // Every column of C equals the matvec result, so take N==0 lanes.
    if (m == 0) {
#pragma unroll
        for (int r = 0; r < 8; ++r)
            outp[row_base + hi * 8 + r] = acc[r];
    }
}

// ---- stage 1: z_partial = W1[tile] * x[chunk] ------------------------------
__global__ void __launch_bounds__(128) fc1_kernel(
    const float* __restrict__ x, const float* __restrict__ W1,
    float* __restrict__ zpart)
{
    __shared__ __align__(16) float xs[2048];
    const int tid = threadIdx.x;
    const int kc  = blockIdx.x & 3;   // K chunk 0..3 (2048 each)
    const int tg  = blockIdx.x >> 2;  // tile group 0..63

    const v4f* x4  = (const v4f*)(x + kc * 2048);
    v4f*       xs4 = (v4f*)xs;
    for (int i = tid; i < 512; i += 128) xs4[i] = x4[i];
    __syncthreads();

    const int wave = tid >> 5;
    const int tile = tg * 4 + wave;   // 0..255
    matvec_tile_wave(W1, IN_DIM, kc * 2048, xs, tile * 16, zpart + kc * H_DIM);
}

// ---- stage 2: z = relu(sum of partials + b1) -------------------------------
__global__ void zrelu_kernel(const float* __restrict__ zpart,
                             const float* __restrict__ b1,
                             float* __restrict__ zbuf)
{
    int i = blockIdx.x * 256 + threadIdx.x;
    float s = zpart[i] + zpart[H_DIM + i] + zpart[2 * H_DIM + i] +
              zpart[3 * H_DIM + i] + b1[i];
    zbuf[i] = s > 0.f ? s : 0.f;
}

// ---- stage 3: gate partials: W_ih * z and W_hh * h0 ------------------------
__global__ void __launch_bounds__(128) gates_kernel(
    const float* __restrict__ zbuf, const float* __restrict__ h0,
    const float* __restrict__ W_ih, const float* __restrict__ W_hh,
    float* __restrict__ gpart)
{
    __shared__ __align__(16) float xs[2048];
    const int tid  = threadIdx.x;
    const int kc   = blockIdx.x & 1;   // K chunk 0..1
    const int tg   = blockIdx.x >> 1;  // 0..511
    const int wave = tid >> 5;
    const int tile = tg * 4 + wave;    // 0..2047 (block-uniform matrix select)
    const int mat  = tile >> 10;       // 0: W_ih (vec=z), 1: W_hh (vec=h0)

    const float* src = mat ? h0 : zbuf;
    const float* W   = mat ? W_hh : W_ih;

    const v4f* s4  = (const v4f*)(src + kc * 2048);
    v4f*       xs4 = (v4f*)xs;
    for (int i = tid; i < 512; i += 128) xs4[i] = s4[i];
    __syncthreads();

    const int row_base = (tile & 1023) * 16;  // gate row 0..16383
    const int slot     = mat * 2 + kc;        // 4 partial slots
    matvec_tile_wave(W, H_DIM, kc * 2048, xs, row_base, gpart + slot * H4_DIM);
}

// ---- stage 4: LSTM cell nonlinearity ---------------------------------------
__global__ void cell_kernel(const float* __restrict__ gpart,
                            const float* __restrict__ b_ih,
                            const float* __restrict__ b_hh,
                            const float* __restrict__ c0,
                            float* __restrict__ out)
{
    int j = blockIdx.x * 256 + threadIdx.x;
    auto gv = [&](int idx) {
        return gpart[idx] + gpart[H4_DIM + idx] + gpart[2 * H4_DIM + idx] +
               gpart[3 * H4_DIM + idx] + b_ih[idx] + b_hh[idx];
    };
    float gi = gv(j);
    float gf = gv(j + H_DIM);
    float gg = gv(j + 2 * H_DIM);
    float go = gv(j + 3 * H_DIM);
    float si = 1.f / (1.f + __expf(-gi));
    float sf = 1.f / (1.f + __expf(-gf));
    float so = 1.f / (1.f + __expf(-go));
    float cn = sf * c0[j] + si * tanhf(gg);
    float hn = so * tanhf(cn);
    out[1 + j]         = hn;  // h_new
    out[1 + H_DIM + j] = cn;  // c_new
}

// ---- stage 5: a = sigmoid(W3 . h + b3); lag blend; y1 ----------------------
__global__ void head_kernel(const float* __restrict__ W3,
                            const float* __restrict__ b3,
                            const float* __restrict__ u_,
                            const float* __restrict__ apre_,
                            const float* __restrict__ ypre_,
                            float* __restrict__ out)
{
    __shared__ float red[256];
    int tid = threadIdx.x;
    const float* h = out + 1;
    float s = 0.f;
    for (int j = tid; j < H_DIM; j += 256) s += W3[j] * h[j];
    red[tid] = s;
    __syncthreads();
    for (int off = 128; off > 0; off >>= 1) {
        if (tid < off) red[tid] += red[tid + off];
        __syncthreads();
    }
    if (tid == 0) {
        float a  = 1.f / (1.f + __expf(-(red[0] + b3[0])));
        float u  = u_[0];
        float ab = u * apre_[0] + (1.f - u) * a;
        float y1 = ab * u + (1.f - ab) * ypre_[0];
        out[0]             = y1;  // y1
        out[1 + 2 * H_DIM] = ab;  // a (blended)
    }
}

// ---------------------------------------------------------------------------
extern "C" void kernel_launch(void* const* d_in, const int* in_sizes, int n_in,
                              void* d_out, int out_size, void* d_ws, size_t ws_size,
                              hipStream_t stream)
{
    const float* x    = (const float*)d_in[0];
    const float* u    = (const float*)d_in[1];
    const float* h0   = (const float*)d_in[2];
    const float* c0   = (const float*)d_in[3];
    const float* ypre = (const float*)d_in[4];
    const float* apre = (const float*)d_in[5];
    const float* W1   = (const float*)d_in[6];
    const float* b1   = (const float*)d_in[7];
    const float* W_ih = (const float*)d_in[8];
    const float* b_ih = (const float*)d_in[9];
    const float* W_hh = (const float*)d_in[10];
    const float* b_hh = (const float*)d_in[11];
    const float* W3   = (const float*)d_in[12];
    const float* b3   = (const float*)d_in[13];

    float* out   = (float*)d_out;
    float* ws    = (float*)d_ws;
    float* zpart = ws + ZPART_OFF;
    float* gpart = ws + GPART_OFF;
    float* zbuf  = ws + ZBUF_OFF;

    fc1_kernel  <<<256,  128, 0, stream>>>(x, W1, zpart);
    zrelu_kernel<<<16,   256, 0, stream>>>(zpart, b1, zbuf);
    gates_kernel<<<1024, 128, 0, stream>>>(zbuf, h0, W_ih, W_hh, gpart);
    cell_kernel <<<16,   256, 0, stream>>>(gpart, b_ih, b_hh, c0, out);
    head_kernel <<<1,    256, 0, stream>>>(W3, b3, u, apre, ypre, out);
}